// MTMSGRU_57088705298489
// MI455X (gfx1250) — compile-verified
//
#include <hip/hip_runtime.h>
#include <math.h>

// ---- model dims -------------------------------------------------------------
#define B_    16
#define HIST_ 8
#define PRED_ 24
#define N_    2000
#define C_    150
#define R_    16
#define F_    11
#define D_    12
#define H_    96
#define KHOP_ 2
#define L_    3
#define SEQ_  32
#define IN_   24
#define S_    32   // H_/L_

typedef __attribute__((ext_vector_type(2))) float v2f;
typedef __attribute__((ext_vector_type(8))) float v8f;

__device__ __forceinline__ v8f wmma4(v2f a, v2f b, v8f c) {
  // V_WMMA_F32_16X16X4_F32 : exact f32 D = A(16x4) x B(4x16) + C
  return __builtin_amdgcn_wmma_f32_16x16x4_f32(false, a, false, b, (short)0, c,
                                               false, false);
}
__device__ __forceinline__ int imind(int a, int b) { return a < b ? a : b; }

// ---------------------------------------------------------------------------
// Batched WMMA GEMM:  Y[b] = op(Amat) @ X[b]   (TRANSA is compile-time)
//   TRANSA=0: Amat row-major (M x Kd), row stride lda.
//   TRANSA=1: Amat physically (Kd x M) row-major, a(m,k)=Amat[k*lda+m].
//   X[b] is (Kd x ncols) row-major (ncols <= 16), Y[b] is (M x ncols).
// 8 waves / block, one 16x16 output tile per wave.  Out-of-range rows/cols
// use clamped loads (garbage only feeds never-stored outputs); only the
// K-tail is explicitly zero-filled.  Requires lda, strideA even (8B align).
// ---------------------------------------------------------------------------
template <int TRANSA>
__global__ __launch_bounds__(256)
void k_wmma_bmm(float* __restrict__ Y, const float* __restrict__ Amat,
                const float* __restrict__ X,
                int M, int Kd, int ncols, int lda,
                long strideA, long strideX, long strideY)
{
  const int b    = blockIdx.z;
  const int wid  = threadIdx.x >> 5;
  const int tile = blockIdx.x * 8 + wid;          // M-tile index (may overrun)
  const int m0   = tile << 4;
  const float* __restrict__ Ab = Amat + (size_t)b * (size_t)strideA;
  const float* __restrict__ Xb = X    + (size_t)b * (size_t)strideX;
  float* __restrict__ Yb       = Y    + (size_t)b * (size_t)strideY;
  const int lane = threadIdx.x & 31;
  const int half = lane >> 4;                     // K{0,1} vs K{2,3}
  const int lr   = lane & 15;
  const int am   = m0 + lr;                       // A-fragment row
  const int amc  = imind(am, M - 1);              // clamped
  const int bn   = lr;                            // output column
  const int bnc  = imind(bn, ncols - 1);          // clamped
  const float* __restrict__ arow =
      TRANSA ? (Ab + amc) : (Ab + (size_t)amc * lda);
  const float* __restrict__ xcol = Xb + bnc;

  v8f acc = {};
  const int Kd4 = Kd & ~3;
  int k0 = 0;
  if (!TRANSA) {
    // blocked main loop: 64 K per iteration, one speculative prefetch,
    // 16 fully-unrolled WMMA chunks (loads get clause-grouped)
    for (; k0 + 64 <= Kd4; k0 += 64) {
      __builtin_prefetch(arow + k0 + 256, 0, 1);  // speculative; overrun OK
#pragma unroll
      for (int kk = 0; kk < 64; kk += 4) {
        const int ka = k0 + kk + 2 * half;
        v2f af = *(const v2f*)(arow + ka);        // global_load_b64
        v2f bf;
        bf.x = xcol[(size_t)ka * ncols];
        bf.y = xcol[(size_t)(ka + 1) * ncols];
        acc = wmma4(af, bf, acc);
      }
    }
  }
  for (; k0 < Kd4; k0 += 4) {                     // generic chunks
    const int ka = k0 + 2 * half;
    v2f af;
    if (TRANSA) {
      af.x = arow[(size_t)ka * lda];
      af.y = arow[(size_t)(ka + 1) * lda];
    } else {
      af = *(const v2f*)(arow + ka);
    }
    v2f bf;
    bf.x = xcol[(size_t)ka * ncols];
    bf.y = xcol[(size_t)(ka + 1) * ncols];
    acc = wmma4(af, bf, acc);
  }
  if (Kd4 < Kd) {                                 // zero-filled K tail
    const int ka = Kd4 + 2 * half;
    float a0 = 0.f, a1 = 0.f, b0 = 0.f, b1 = 0.f;
    if (ka < Kd) {
      a0 = TRANSA ? arow[(size_t)ka * lda] : arow[ka];
      b0 = xcol[(size_t)ka * ncols];
    }
    if (ka + 1 < Kd) {
      a1 = TRANSA ? arow[(size_t)(ka + 1) * lda] : arow[ka + 1];
      b1 = xcol[(size_t)(ka + 1) * ncols];
    }
    v2f af; af.x = a0; af.y = a1;
    v2f bf; bf.x = b0; bf.y = b1;
    acc = wmma4(af, bf, acc);
  }
#pragma unroll
  for (int v = 0; v < 8; ++v) {
    const int m = m0 + v + 8 * half;              // C layout: VGPR v -> row v(+8)
    if (m < M && bn < ncols) Yb[(size_t)m * ncols + bn] = acc[v];
  }
}

// ---------------------------------------------------------------------------
// WMMA GEMM:  Y = X @ W^T + bias   (the _lin pattern)
//   X (M x Kd) row-major, W (Ncols x Kd) row-major, Y (M x Ncols).
// 8 waves / block; each wave computes a 16(M) x 32(N) strip: two accumulators
// sharing one A fragment per K-chunk.  Requires Kd even (8B-aligned rows).
// ---------------------------------------------------------------------------
__global__ __launch_bounds__(256)
void k_wmma_lin(float* __restrict__ Y, const float* __restrict__ X,
                const float* __restrict__ W, const float* __restrict__ bias,
                int M, int Kd, int Ncols)
{
  const int nT2  = (Ncols + 31) >> 5;             // N strips of 32
  const int wid  = threadIdx.x >> 5;
  const int flat = blockIdx.x * 8 + wid;
  const int mt   = flat / nT2;
  const int nt   = flat % nT2;
  const int m0   = mt << 4;
  const int n0   = nt << 5;
  const int lane = threadIdx.x & 31;
  const int half = lane >> 4;
  const int lr   = lane & 15;
  const int am   = m0 + lr;
  const int amc  = imind(am, M - 1);
  const int bn0  = n0 + lr;
  const int bn1  = n0 + 16 + lr;
  const int b0c  = imind(bn0, Ncols - 1);
  const int b1c  = imind(bn1, Ncols - 1);
  const float* __restrict__ xrow = X + (size_t)amc * Kd;
  const float* __restrict__ w0   = W + (size_t)b0c * Kd;
  const float* __restrict__ w1   = W + (size_t)b1c * Kd;

  v8f acc0 = {}, acc1 = {};
  const int Kd4 = Kd & ~3;
#pragma unroll 4
  for (int k0 = 0; k0 < Kd4; k0 += 4) {
    const int ka = k0 + 2 * half;
    v2f af  = *(const v2f*)(xrow + ka);           // global_load_b64
    v2f bf0 = *(const v2f*)(w0 + ka);
    v2f bf1 = *(const v2f*)(w1 + ka);
    acc0 = wmma4(af, bf0, acc0);
    acc1 = wmma4(af, bf1, acc1);
  }
  if (Kd4 < Kd) {                                 // zero-filled K tail
    const int ka = Kd4 + 2 * half;
    float a0 = 0.f, a1 = 0.f, c0 = 0.f, c1 = 0.f, d0 = 0.f, d1 = 0.f;
    if (ka < Kd)     { a0 = xrow[ka];     c0 = w0[ka];     d0 = w1[ka]; }
    if (ka + 1 < Kd) { a1 = xrow[ka + 1]; c1 = w0[ka + 1]; d1 = w1[ka + 1]; }
    v2f af; af.x = a0; af.y = a1;
    v2f bf0; bf0.x = c0; bf0.y = c1;
    v2f bf1; bf1.x = d0; bf1.y = d1;
    acc0 = wmma4(af, bf0, acc0);
    acc1 = wmma4(af, bf1, acc1);
  }
  const float bv0 = (bn0 < Ncols) ? bias[bn0] : 0.f;
  const float bv1 = (bn1 < Ncols) ? bias[bn1] : 0.f;
#pragma unroll
  for (int v = 0; v < 8; ++v) {
    const int m = m0 + v + 8 * half;
    if (m < M) {
      if (bn0 < Ncols) Y[(size_t)m * Ncols + bn0] = acc0[v] + bv0;
      if (bn1 < Ncols) Y[(size_t)m * Ncols + bn1] = acc1[v] + bv1;
    }
  }
}

// ---------------------------------------------------------------------------
// small / elementwise kernels
// ---------------------------------------------------------------------------
__global__ void k_zero(float* p, long n) {
  long i = (long)blockIdx.x * blockDim.x + threadIdx.x;
  if (i < n) p[i] = 0.f;
}

// afc_c2r[b,r,c] = softmax_r( sum_h sc_fc_w[h]*(row_W[r,:] . cin[b,h,c,:]) + sc_fc_b )
__global__ void k_afc(float* afc, const float* c_x_hist, const float* city_info,
                      const float* row_W, const float* sc_fc_w, const float* sc_fc_b)
{
  int i = blockIdx.x * blockDim.x + threadIdx.x;
  if (i >= B_ * C_) return;
  int b = i / C_, c = i % C_;
  float ci0 = city_info[c * 2 + 0], ci1 = city_info[c * 2 + 1];
  float s0 = 0.f, s1 = 0.f;
  for (int h = 0; h < HIST_; ++h) {
    float w = sc_fc_w[h];
    s0 += w;
    s1 += w * c_x_hist[((size_t)b * HIST_ + h) * C_ + c];
  }
  float val[R_]; float mx = -1e30f;
  for (int r = 0; r < R_; ++r) {
    float v = s0 * (row_W[r * 3 + 0] * ci0 + row_W[r * 3 + 1] * ci1)
            + row_W[r * 3 + 2] * s1 + sc_fc_b[0];
    val[r] = v; mx = fmaxf(mx, v);
  }
  float sum = 0.f;
  for (int r = 0; r < R_; ++r) { val[r] = __expf(val[r] - mx); sum += val[r]; }
  float inv = 1.f / sum;
  for (int r = 0; r < R_; ++r) afc[((size_t)b * R_ + r) * C_ + c] = val[r] * inv;
}

// out[b,t,r,f] = sum_c afc[b,r,c] * in[b,t,c,f]
__global__ void k_c2r_apply(float* out, const float* in, const float* afc,
                            int T, int Fdim)
{
  long i = (long)blockIdx.x * blockDim.x + threadIdx.x;
  long total = (long)B_ * T * R_ * Fdim;
  if (i >= total) return;
  int f = (int)(i % Fdim); long t1 = i / Fdim;
  int r = (int)(t1 % R_);  long t2 = t1 / R_;
  int t = (int)(t2 % T);   int b = (int)(t2 / T);
  const float* arow = afc + ((size_t)b * R_ + r) * C_;
  const float* inb  = in + ((size_t)b * T + t) * C_ * Fdim;
  float s = 0.f;
  for (int c = 0; c < C_; ++c) s += arow[c] * inb[(size_t)c * Fdim + f];
  out[i] = s;
}

// global wind stats over rfeat(B,SEQ,R,F) cols 1..2 -> scale = 1 + 0.1*mean(tanh(m/(sd+1e-5)))
__global__ void k_scale(float* scale_out, const float* rfeat)
{
  __shared__ float sh[1024];
  int tid = threadIdx.x;
  long nrows = (long)B_ * SEQ_ * R_;
  float s1 = 0, s2 = 0, q1 = 0, q2 = 0;
  for (long row = tid; row < nrows; row += blockDim.x) {
    float v1 = rfeat[row * F_ + 1];
    float v2 = rfeat[row * F_ + 2];
    s1 += v1; s2 += v2; q1 += v1 * v1; q2 += v2 * v2;
  }
  sh[tid] = s1; sh[256 + tid] = s2; sh[512 + tid] = q1; sh[768 + tid] = q2;
  __syncthreads();
  for (int st = 128; st > 0; st >>= 1) {
    if (tid < st) {
      sh[tid] += sh[tid + st]; sh[256 + tid] += sh[256 + tid + st];
      sh[512 + tid] += sh[512 + tid + st]; sh[768 + tid] += sh[768 + tid + st];
    }
    __syncthreads();
  }
  if (tid == 0) {
    float n = (float)nrows;
    float m1 = sh[0] / n, m2 = sh[256] / n;
    float v1 = (sh[512] - n * m1 * m1) / (n - 1.f);
    float v2 = (sh[768] - n * m2 * m2) / (n - 1.f);
    float sd1 = sqrtf(fmaxf(v1, 0.f)), sd2 = sqrtf(fmaxf(v2, 0.f));
    float s = 0.5f * (tanhf(m1 / (sd1 + 1e-5f)) + tanhf(m2 / (sd2 + 1e-5f)));
    scale_out[0] = 1.0f + 0.1f * s;
  }
}

// adj[b,r,q] = softmax_q( (afc[b,r,:].dist_w[q,:]+dist_b[q]) * sigmoid(direc...) )
__global__ void k_adj(float* adj, const float* afc,
                      const float* dist_w, const float* dist_b,
                      const float* direc_w, const float* direc_b)
{
  int i = blockIdx.x * blockDim.x + threadIdx.x;
  if (i >= B_ * R_) return;
  int b = i / R_, r = i % R_;
  const float* arow = afc + ((size_t)b * R_ + r) * C_;
  float lg[R_]; float mx = -1e30f;
  for (int q = 0; q < R_; ++q) {
    float da = dist_b[q], dc = direc_b[q];
    const float* dw = dist_w + (size_t)q * C_;
    const float* cw = direc_w + (size_t)q * C_;
    for (int c = 0; c < C_; ++c) { da += arow[c] * dw[c]; dc += arow[c] * cw[c]; }
    float v = da * (1.f / (1.f + __expf(-dc)));
    lg[q] = v; mx = fmaxf(mx, v);
  }
  float sum = 0.f;
  for (int q = 0; q < R_; ++q) { lg[q] = __expf(lg[q] - mx); sum += lg[q]; }
  float inv = 1.f / sum;
  for (int q = 0; q < R_; ++q) adj[((size_t)b * R_ + r) * R_ + q] = lg[q] * inv;
}

// y = tanh(T @ W + bias) [* scale]   (W used non-transposed, _gnn pattern)
__global__ void k_stage2_gnn(float* y, const float* T, const float* W,
                             const float* bias, long rows, const float* scale)
{
  long i = (long)blockIdx.x * blockDim.x + threadIdx.x;
  if (i >= rows * D_) return;
  long row = i / D_; int j = (int)(i % D_);
  const float* tr = T + row * D_;
  float s = bias[j];
  for (int d = 0; d < D_; ++d) s += tr[d] * W[d * D_ + j];
  s = tanhf(s);
  if (scale) s *= scale[0];
  y[i] = s;
}

// y += T @ W^T + bias   (_lin pattern)
__global__ void k_stage2_addlin(float* y, const float* T, const float* W,
                                const float* bias, long rows)
{
  long i = (long)blockIdx.x * blockDim.x + threadIdx.x;
  if (i >= rows * D_) return;
  long row = i / D_; int j = (int)(i % D_);
  const float* tr = T + row * D_;
  float s = bias[j];
  for (int d = 0; d < D_; ++d) s += tr[d] * W[j * D_ + d];
  y[i] += s;
}

// xcat[row, 0:12] = xg[row], xcat[row, 12:24] = xn[row]
__global__ void k_concat(float* xcat, const float* xg, const float* xn, long rows)
{
  long i = (long)blockIdx.x * blockDim.x + threadIdx.x;
  if (i >= rows * IN_) return;
  long row = i / IN_; int c = (int)(i % IN_);
  xcat[i] = (c < D_) ? xg[row * D_ + c] : xn[row * D_ + (c - D_)];
}

// attention a = softmax_l(x@x2a^T + h@h2a^T + b), then wh = h * a[chunk]
__global__ void k_attn_wh(float* __restrict__ wh, const float* __restrict__ xcat,
                          const float* __restrict__ h,
                          const float* x2a_w, const float* x2a_b,
                          const float* h2a_w, const float* h2a_b, int rows)
{
  int row = blockIdx.x * blockDim.x + threadIdx.x;
  if (row >= rows) return;
  const float* xr = xcat + (size_t)row * IN_;
  const float* hr = h + (size_t)row * H_;
  float lg[L_];
  for (int l = 0; l < L_; ++l) {
    float s = x2a_b[l] + h2a_b[l];
    const float* wx = x2a_w + l * IN_;
    for (int i = 0; i < IN_; ++i) s += xr[i] * wx[i];
    const float* wv = h2a_w + l * H_;
    for (int j = 0; j < H_; ++j) s += hr[j] * wv[j];
    lg[l] = s;
  }
  float mx = fmaxf(lg[0], fmaxf(lg[1], lg[2]));
  float e0 = __expf(lg[0] - mx), e1 = __expf(lg[1] - mx), e2 = __expf(lg[2] - mx);
  float inv = 1.f / (e0 + e1 + e2);
  float a0 = e0 * inv, a1 = e1 * inv, a2 = e2 * inv;
  float* wr = wh + (size_t)row * H_;
  for (int j = 0; j < H_; ++j) wr[j] = hr[j] * ((j < S_) ? a0 : (j < 2 * S_) ? a1 : a2);
}

// GRU gate fusion + interval-masked in-place state update
__global__ void k_gru_final(float* __restrict__ h, const float* gx, const float* gh,
                            const float* wh, int rows, int idx)
{
  long i = (long)blockIdx.x * blockDim.x + threadIdx.x;
  if (i >= (long)rows * H_) return;
  long row = i / H_; int col = (int)(i % H_);
  size_t g = (size_t)row * 3 * H_;
  float xz = gx[g + col], xr = gx[g + H_ + col], xn = gx[g + 2 * H_ + col];
  float hz = gh[g + col], hr = gh[g + H_ + col], hn = gh[g + 2 * H_ + col];
  float rg = 1.f / (1.f + __expf(-(xr + hr)));
  float zg = 1.f / (1.f + __expf(-(xz + hz)));
  float ng = tanhf(xn + rg * hn);
  float w = wh[(size_t)row * H_ + col];
  float upd = ng + zg * (w - ng);
  int chunk = col / S_;
  int interval = (chunk == 0) ? 1 : ((chunk == 1) ? 3 : 6);
  if (idx % interval == 0) h[(size_t)row * H_ + col] = upd;
}

// x = (h0+h1) @ w^T + b ; optionally scatter into d_out at (b, step, n)
__global__ void k_fcout(float* __restrict__ xbuf, float* __restrict__ outp,
                        const float* h0, const float* h1,
                        const float* w, const float* bias,
                        int rows, int ncb, int step)
{
  int row = blockIdx.x * blockDim.x + threadIdx.x;
  if (row >= rows) return;
  const float* a = h0 + (size_t)row * H_;
  const float* bb = h1 + (size_t)row * H_;
  float s = bias[0];
  for (int j = 0; j < H_; ++j) s += (a[j] + bb[j]) * w[j];
  xbuf[row] = s;
  if (outp) {
    int b = row / ncb, n = row % ncb;
    outp[((size_t)b * PRED_ + step) * ncb + n] = s;
  }
}

// build step input xn (and copy of it into xg) from history tensors
__global__ void k_build_hist(float* xn, float* xg, const float* xsrc,
                             const float* fsrc, int t, int ncb, int Tx)
{
  int i = blockIdx.x * blockDim.x + threadIdx.x;
  if (i >= B_ * ncb) return;
  int b = i / ncb, n = i % ncb;
  float xv = xsrc[((size_t)b * Tx + t) * ncb + n];
  size_t base = (size_t)i * D_;
  xn[base] = xv; xg[base] = xv;
  const float* fr = fsrc + (((size_t)b * SEQ_ + t) * ncb + n) * F_;
  for (int f = 0; f < F_; ++f) { float v = fr[f]; xn[base + 1 + f] = v; xg[base + 1 + f] = v; }
}

// build step input from recurrent x buffer + features at time tf
__global__ void k_build_pred(float* xn, float* xg, const float* xbuf,
                             const float* fsrc, int tf, int ncb)
{
  int i = blockIdx.x * blockDim.x + threadIdx.x;
  if (i >= B_ * ncb) return;
  int b = i / ncb, n = i % ncb;
  float xv = xbuf[i];
  size_t base = (size_t)i * D_;
  xn[base] = xv; xg[base] = xv;
  const float* fr = fsrc + (((size_t)b * SEQ_ + tf) * ncb + n) * F_;
  for (int f = 0; f < F_; ++f) { float v = fr[f]; xn[base + 1 + f] = v; xg[base + 1 + f] = v; }
}

__global__ void k_init_x(float* xbuf, const float* xsrc, int ncb, int Tx, int t)
{
  int i = blockIdx.x * blockDim.x + threadIdx.x;
  if (i >= B_ * ncb) return;
  int b = i / ncb, n = i % ncb;
  xbuf[i] = xsrc[((size_t)b * Tx + t) * ncb + n];
}

// ---------------------------------------------------------------------------
// host orchestration
// ---------------------------------------------------------------------------
struct GruW {
  const float *x2h_w, *x2h_b, *h2h_w, *h2h_b, *x2a_w, *x2a_b, *h2a_w, *h2a_b;
};
static GruW load_gru(void* const* d_in, int base) {
  GruW g;
  g.x2h_w = (const float*)d_in[base + 0];
  g.x2h_b = (const float*)d_in[base + 1];
  g.h2h_w = (const float*)d_in[base + 2];
  g.h2h_b = (const float*)d_in[base + 3];
  g.x2a_w = (const float*)d_in[base + 4];
  g.x2a_b = (const float*)d_in[base + 5];
  g.h2a_w = (const float*)d_in[base + 6];
  g.h2a_b = (const float*)d_in[base + 7];
  return g;
}
static inline dim3 g1(long n) { return dim3((unsigned)((n + 255) / 256)); }

extern "C" void kernel_launch(void* const* d_in, const int* in_sizes, int n_in,
                              void* d_out, int out_size, void* d_ws, size_t ws_size,
                              hipStream_t stream)
{
  (void)in_sizes; (void)n_in; (void)out_size; (void)ws_size;
  const float* x_hist     = (const float*)d_in[0];
  const float* features   = (const float*)d_in[1];
  const float* c_x_hist   = (const float*)d_in[2];
  const float* c_features = (const float*)d_in[3];
  const float* city_info  = (const float*)d_in[4];
  const float* afc_s2c    = (const float*)d_in[5];
  const float* row_W      = (const float*)d_in[6];
  const float* sc_fc_w    = (const float*)d_in[7];
  const float* sc_fc_b    = (const float*)d_in[8];
  const float* dist_w     = (const float*)d_in[9];
  const float* dist_b     = (const float*)d_in[10];
  const float* direc_w    = (const float*)d_in[11];
  const float* direc_b    = (const float*)d_in[12];
  const float* Amat       = (const float*)d_in[13];
  const float* gnn_w      = (const float*)d_in[14];
  const float* gnn_b      = (const float*)d_in[15];
  const float* Ac         = (const float*)d_in[16];
  const float* c_gnn_w    = (const float*)d_in[17];
  const float* c_gnn_b    = (const float*)d_in[18];
  const float* r_gnn_w    = (const float*)d_in[19];
  const float* r_gnn_b    = (const float*)d_in[20];
  const float* c2s_w      = (const float*)d_in[21];
  const float* c2s_b      = (const float*)d_in[22];
  const float* s2c_w      = (const float*)d_in[23];
  const float* s2c_b      = (const float*)d_in[24];
  const float* r2c_w      = (const float*)d_in[25];
  const float* r2c_b      = (const float*)d_in[26];
  const float* fc_out_w   = (const float*)d_in[27];
  const float* fc_out_b   = (const float*)d_in[28];
  const float* c_fc_out_w = (const float*)d_in[29];
  const float* c_fc_out_b = (const float*)d_in[30];
  const float* r_fc_out_w = (const float*)d_in[31];
  const float* r_fc_out_b = (const float*)d_in[32];
  GruW gs = load_gru(d_in, 33);
  GruW gc = load_gru(d_in, 41);
  GruW gr = load_gru(d_in, 49);

  float* outY = (float*)d_out;                        // (B,PRED,N,1)
  float* outC = outY + (size_t)B_ * PRED_ * N_;       // (B,PRED,C,1)

  // ---- workspace carve ----
  size_t off = 0; char* base = (char*)d_ws;
  auto alloc = [&](size_t elems) -> float* {
    float* p = (float*)(base + off);
    off += ((elems * sizeof(float)) + 255) & ~(size_t)255;
    return p;
  };
  float* p_afc  = alloc((size_t)B_ * R_ * C_);
  float* p_rxh  = alloc((size_t)B_ * HIST_ * R_);
  float* p_rft  = alloc((size_t)B_ * SEQ_ * R_ * F_);
  float* p_adj  = alloc((size_t)B_ * R_ * R_);
  float* p_scl  = alloc(8);
  float* p_xg   = alloc((size_t)B_ * N_ * D_);
  float* p_xcg  = alloc((size_t)B_ * C_ * D_);
  float* p_xrg  = alloc((size_t)B_ * R_ * D_);
  float* p_xn   = alloc((size_t)B_ * N_ * D_);
  float* p_xcn  = alloc((size_t)B_ * C_ * D_);
  float* p_xrn  = alloc((size_t)B_ * R_ * D_);
  float* p_Tn   = alloc((size_t)B_ * N_ * D_);   // big stage-1 temp (reused)
  float* p_Tc   = alloc((size_t)B_ * C_ * D_);   // city stage-1 temp (reused)
  float* p_Tr   = alloc((size_t)B_ * R_ * D_);
  float* p_T2   = alloc((size_t)B_ * C_ * D_);
  float* p_cat  = alloc((size_t)B_ * N_ * IN_);  // shared across levels (serial)
  float* p_wh   = alloc((size_t)B_ * N_ * H_);
  float* p_gx   = alloc((size_t)B_ * N_ * 3 * H_);
  float* p_gh   = alloc((size_t)B_ * N_ * 3 * H_);
  long   hszS = (long)B_ * N_ * H_, hszC = (long)B_ * C_ * H_, hszR = (long)B_ * R_ * H_;
  float* p_hS   = alloc((size_t)KHOP_ * hszS);
  float* p_hC   = alloc((size_t)KHOP_ * hszC);
  float* p_hR   = alloc((size_t)KHOP_ * hszR);
  float* p_xbS  = alloc((size_t)B_ * N_);
  float* p_xbC  = alloc((size_t)B_ * C_);
  float* p_xbR  = alloc((size_t)B_ * R_);

  // ---- init hidden states to zero (required every call) ----
  k_zero<<<g1(KHOP_ * hszS), 256, 0, stream>>>(p_hS, KHOP_ * hszS);
  k_zero<<<g1(KHOP_ * hszC), 256, 0, stream>>>(p_hC, KHOP_ * hszC);
  k_zero<<<g1(KHOP_ * hszR), 256, 0, stream>>>(p_hR, KHOP_ * hszR);

  // ---- precompute ----
  k_afc<<<g1(B_ * C_), 256, 0, stream>>>(p_afc, c_x_hist, city_info, row_W, sc_fc_w, sc_fc_b);
  k_c2r_apply<<<g1((long)B_ * SEQ_ * R_ * F_), 256, 0, stream>>>(p_rft, c_features, p_afc, SEQ_, F_);
  k_c2r_apply<<<g1((long)B_ * HIST_ * R_ * 1), 256, 0, stream>>>(p_rxh, c_x_hist, p_afc, HIST_, 1);
  k_scale<<<1, 256, 0, stream>>>(p_scl, p_rft);
  k_adj<<<g1(B_ * R_), 256, 0, stream>>>(p_adj, p_afc, dist_w, dist_b, direc_w, direc_b);

  // tiles: M-tiles per batched GEMM, packed 8 waves/block
  const int tS = ((N_ + 15) / 16 + 7) / 8;   // station: 125 tiles -> 16 blocks
  const int tC = ((C_ + 15) / 16 + 7) / 8;   // city: 10 tiles -> 2 blocks
  const int tR = 1;                          // region: 1 tile
  const long rS = (long)B_ * N_, rC = (long)B_ * C_, rR = (long)B_ * R_;
  const int nT2 = (3 * H_ + 31) / 32;        // 9 N-strips of 32 for the gates
  const int mS = (int)((rS + 15) / 16), mC = (int)((rC + 15) / 16), mR = (int)((rR + 15) / 16);
  const int gS = (mS * nT2 + 7) / 8, gC = (mC * nT2 + 7) / 8, gR = (mR * nT2 + 7) / 8;

  auto cell = [&](const GruW& g, float* xgv, const float* xnv,
                  float* h, long rows, int gridLin, int k, int idx) {
    k_concat<<<g1(rows * IN_), 256, 0, stream>>>(p_cat, xgv, xnv, rows);
    k_attn_wh<<<g1(rows), 256, 0, stream>>>(p_wh, p_cat, h,
        g.x2a_w + (size_t)k * L_ * IN_, g.x2a_b + (size_t)k * L_,
        g.h2a_w + (size_t)k * L_ * H_,  g.h2a_b + (size_t)k * L_, (int)rows);
    k_wmma_lin<<<dim3(gridLin), 256, 0, stream>>>(p_gx, p_cat,
        g.x2h_w + (size_t)k * 3 * H_ * IN_, g.x2h_b + (size_t)k * 3 * H_,
        (int)rows, IN_, 3 * H_);
    k_wmma_lin<<<dim3(gridLin), 256, 0, stream>>>(p_gh, p_wh,
        g.h2h_w + (size_t)k * 3 * H_ * H_, g.h2h_b + (size_t)k * 3 * H_,
        (int)rows, H_, 3 * H_);
    k_gru_final<<<g1(rows * H_), 256, 0, stream>>>(h, p_gx, p_gh, p_wh, (int)rows, idx);
  };

  int idx = 1;
  for (int t = 0; t < (HIST_ - 1) + PRED_; ++t) {
    const bool hist = (t < HIST_ - 1);
    if (hist) {
      k_build_hist<<<g1(B_ * N_), 256, 0, stream>>>(p_xn, p_xg, x_hist, features, t, N_, HIST_);
      k_build_hist<<<g1(B_ * C_), 256, 0, stream>>>(p_xcn, p_xcg, c_x_hist, c_features, t, C_, HIST_);
      k_build_hist<<<g1(B_ * R_), 256, 0, stream>>>(p_xrn, p_xrg, p_rxh, p_rft, t, R_, HIST_);
    } else {
      if (t == HIST_ - 1) { // x/xc/xr <- last history frame
        k_init_x<<<g1(B_ * N_), 256, 0, stream>>>(p_xbS, x_hist, N_, HIST_, HIST_ - 1);
        k_init_x<<<g1(B_ * C_), 256, 0, stream>>>(p_xbC, c_x_hist, C_, HIST_, HIST_ - 1);
        k_init_x<<<g1(B_ * R_), 256, 0, stream>>>(p_xbR, p_rxh, R_, HIST_, HIST_ - 1);
      }
      const int tf = t; // HIST-1 + j
      k_build_pred<<<g1(B_ * N_), 256, 0, stream>>>(p_xn, p_xg, p_xbS, features, tf, N_);
      k_build_pred<<<g1(B_ * C_), 256, 0, stream>>>(p_xcn, p_xcg, p_xbC, c_features, tf, C_);
      k_build_pred<<<g1(B_ * R_), 256, 0, stream>>>(p_xrn, p_xrg, p_xbR, p_rft, tf, R_);
    }

    for (int k = 0; k < KHOP_; ++k) {
      // station GNN: T = A @ xg ; xg = tanh(T @ gnn_w + b)
      k_wmma_bmm<0><<<dim3(tS, 1, B_), 256, 0, stream>>>(p_Tn, Amat, p_xg,
          N_, N_, D_, N_, 0L, (long)N_ * D_, (long)N_ * D_);
      k_stage2_gnn<<<g1(rS * D_), 256, 0, stream>>>(p_xg, p_Tn, gnn_w, gnn_b, rS, nullptr);
      // city GNN
      k_wmma_bmm<0><<<dim3(tC, 1, B_), 256, 0, stream>>>(p_Tc, Ac, p_xcg,
          C_, C_, D_, C_, 0L, (long)C_ * D_, (long)C_ * D_);
      k_stage2_gnn<<<g1(rC * D_), 256, 0, stream>>>(p_xcg, p_Tc, c_gnn_w, c_gnn_b, rC, nullptr);
      // region GNN (per-batch adj) with scale
      k_wmma_bmm<0><<<dim3(tR, 1, B_), 256, 0, stream>>>(p_Tr, p_adj, p_xrg,
          R_, R_, D_, R_, (long)R_ * R_, (long)R_ * D_, (long)R_ * D_);
      k_stage2_gnn<<<g1(rR * D_), 256, 0, stream>>>(p_xrg, p_Tr, r_gnn_w, r_gnn_b, rR, p_scl);
      // r2c: xcg += lin(afc^T @ xrg)
      k_wmma_bmm<1><<<dim3(tC, 1, B_), 256, 0, stream>>>(p_Tc, p_afc, p_xrg,
          C_, R_, D_, C_, (long)R_ * C_, (long)R_ * D_, (long)C_ * D_);
      k_stage2_addlin<<<g1(rC * D_), 256, 0, stream>>>(p_xcg, p_Tc, r2c_w, r2c_b, rC);
      // s2c stage1 BEFORE xg is updated (uses tmp = post-GNN xg)
      k_wmma_bmm<1><<<dim3(tC, 1, B_), 256, 0, stream>>>(p_T2, afc_s2c, p_xg,
          C_, N_, D_, C_, 0L, (long)N_ * D_, (long)C_ * D_);
      // c2s: xg += lin(afc_s2c @ xcg)
      k_wmma_bmm<0><<<dim3(tS, 1, B_), 256, 0, stream>>>(p_Tn, afc_s2c, p_xcg,
          N_, C_, D_, C_, 0L, (long)C_ * D_, (long)N_ * D_);
      k_stage2_addlin<<<g1(rS * D_), 256, 0, stream>>>(p_xg, p_Tn, c2s_w, c2s_b, rS);
      // s2c stage2: xcg += lin(T2)
      k_stage2_addlin<<<g1(rC * D_), 256, 0, stream>>>(p_xcg, p_T2, s2c_w, s2c_b, rC);

      // GRU cells (serial; share big cat/wh/gx/gh scratch)
      cell(gs, p_xg,  p_xn,  p_hS + (size_t)k * hszS, rS, gS, k, idx);
      cell(gc, p_xcg, p_xcn, p_hC + (size_t)k * hszC, rC, gC, k, idx);
      cell(gr, p_xrg, p_xrn, p_hR + (size_t)k * hszR, rR, gR, k, idx);
      ++idx;
    }

    if (!hist) {
      const int j = t - (HIST_ - 1);
      k_fcout<<<g1(rS), 256, 0, stream>>>(p_xbS, outY, p_hS, p_hS + hszS,
                                          fc_out_w, fc_out_b, (int)rS, N_, j);
      k_fcout<<<g1(rC), 256, 0, stream>>>(p_xbC, outC, p_hC, p_hC + hszC,
                                          c_fc_out_w, c_fc_out_b, (int)rC, C_, j);
      k_fcout<<<g1(rR), 256, 0, stream>>>(p_xbR, nullptr, p_hR, p_hR + hszR,
                                          r_fc_out_w, r_fc_out_b, (int)rR, R_, j);
    }
  }
}